// ComponentChangeEncoder_65335042507172
// MI455X (gfx1250) — compile-verified
//
#include <hip/hip_runtime.h>
#include <math.h>

typedef float v2f __attribute__((ext_vector_type(2)));
typedef float v8f __attribute__((ext_vector_type(8)));
typedef unsigned int v4u __attribute__((ext_vector_type(4)));
typedef unsigned int v8u __attribute__((ext_vector_type(8)));

#define BB   256
#define NN   512
#define BNR  (BB*NN)          // 131072 rows per set
#define FEAT 512
#define HID  1024

__device__ __forceinline__ v8f wmma4(v2f a, v2f b, v8f c) {
    // V_WMMA_F32_16X16X4_F32 : D = A(16x4) * B(4x16) + C(16x16), fp32 exact
    return __builtin_amdgcn_wmma_f32_16x16x4_f32(
        /*neg_a=*/false, a, /*neg_b=*/false, b,
        /*c_mod=*/(short)0, c, /*reuse_a=*/false, /*reuse_b=*/false);
}

// ---------------------------------------------------------------------------
// Tensor Data Mover: async 2D tile load, global -> LDS.
//   rows x tile_k fp32 elements, row stride row_stride_elems in memory.
//   LDS padding: +2 dwords after every 256 dwords -> LDS row stride 516 words
//   (516 mod 64 == 4 -> conflict-free ds_load_b64 fragment reads).
// Inline asm (not the clang builtin) so the code is portable across the
// 5-arg (ROCm 7.2) / 6-arg (therock) builtin signatures.
// ---------------------------------------------------------------------------
__device__ __forceinline__ void tdm_load_2d(unsigned int lds_byte_off,
                                            unsigned long long gaddr,
                                            unsigned int tile_k,
                                            unsigned int rows,
                                            unsigned int row_stride_elems)
{
    v4u g0;
    g0[0] = 1u;                                        // count=1, user descriptor
    g0[1] = lds_byte_off;                              // lds_addr (bytes)
    g0[2] = (unsigned int)(gaddr & 0xFFFFFFFFull);     // global_addr[31:0]
    g0[3] = (unsigned int)((gaddr >> 32) & 0x01FFFFFFull) | (2u << 30); // [56:32] | type=2
    v8u g1;
    g1[0] = (2u << 16)          // data_size = 4B
          | (1u << 20)          // pad_enable
          | (7u << 22)          // pad_interval code 7 = 256 dwords
          | (1u << 25);         // pad_amount  code 1 = 2 dwords
    g1[1] = (tile_k & 0xFFFFu) << 16;                  // tensor_dim0[15:0] @ [63:48]
    g1[2] = (rows   & 0xFFFFu) << 16;                  // tensor_dim1[15:0] @ [95:80]
    g1[3] = (tile_k & 0xFFFFu) << 16;                  // tile_dim0 @ [127:112]
    g1[4] = rows & 0xFFFFu;                            // tile_dim1 @ [143:128], tile_dim2=0
    g1[5] = row_stride_elems;                          // tensor_dim0_stride[31:0]
    g1[6] = 0u;
    g1[7] = 0u;
    v4u gz = {0u, 0u, 0u, 0u};                         // groups 2/3 unused (2-D tile)
    asm volatile("tensor_load_to_lds %0, %1, %2, %3"
                 :: "s"(g0), "s"(g1), "s"(gz), "s"(gz)
                 : "memory");
}

// ---------------------------------------------------------------------------
// Encoder: rows 0..BNR-1 = ref, BNR..2*BNR-1 = tar. One wave per 16-row tile.
// ---------------------------------------------------------------------------
__global__ __launch_bounds__(32) void encode_kernel(
    const float* __restrict__ refc, const float* __restrict__ tarc,
    const float* __restrict__ temb,
    const float* __restrict__ w1, const float* __restrict__ b1,
    const float* __restrict__ g1, const float* __restrict__ be1,
    const float* __restrict__ w2, const float* __restrict__ b2,
    const float* __restrict__ g2, const float* __restrict__ be2,
    const float* __restrict__ w3, const float* __restrict__ b3,
    float* __restrict__ out_ref, float* __restrict__ out_tar)
{
    __shared__ float X[16*44];       // input 44-dim rows
    __shared__ float H1[16*256];
    __shared__ float H2[16*512];
    __shared__ float MEANS[16];
    __shared__ float RSTDS[16];
    __shared__ float TYPEF[16];

    const int lane = threadIdx.x & 31;
    const int half = lane >> 4;
    const int ml   = lane & 15;
    const int gRow = blockIdx.x * 16;
    const bool isRef = (gRow < BNR);
    const float* src = isRef ? refc : tarc;
    const int rbase = isRef ? gRow : gRow - BNR;

    // stage types
    if (lane < 16) {
        int t = (int)src[(size_t)(rbase + lane) * 13];
        t = t < 0 ? 0 : (t > 19 ? 19 : t);
        TYPEF[lane] = (float)t;
    }
    __syncthreads();

    // stage x = [emb(32) | feats(12)]  (16*44 = 704 = 22*32 exact)
    for (int li = lane; li < 16 * 44; li += 32) {
        const int row = li / 44, c = li % 44;
        const int t = (int)TYPEF[row];
        float v = (c < 32) ? temb[t * 32 + c]
                           : src[(size_t)(rbase + row) * 13 + 1 + (c - 32)];
        X[row * 44 + c] = v;
    }
    __syncthreads();

    // ---- layer 1: 44 -> 256 ----
    for (int nt = 0; nt < 16; ++nt) {
        const int n0 = nt * 16;
        v8f acc = {0.f,0.f,0.f,0.f,0.f,0.f,0.f,0.f};
        for (int k0 = 0; k0 < 44; k0 += 4) {
            const int kk = k0 + 2 * half;
            v2f a; a.x = X[ml * 44 + kk]; a.y = X[ml * 44 + kk + 1];
            v2f bf; bf.x = w1[kk * 256 + n0 + ml]; bf.y = w1[(kk + 1) * 256 + n0 + ml];
            acc = wmma4(a, bf, acc);
        }
        const float bbv = b1[n0 + ml];
        for (int r = 0; r < 8; ++r)
            H1[(r + 8 * half) * 256 + n0 + ml] = acc[r] + bbv;
    }
    __syncthreads();
    if (lane < 16) {
        float s = 0.f, s2 = 0.f;
        const float* row = &H1[lane * 256];
        for (int c = 0; c < 256; ++c) { float v = row[c]; s += v; s2 += v * v; }
        const float m = s * (1.f / 256.f);
        MEANS[lane] = m;
        RSTDS[lane] = rsqrtf(s2 * (1.f / 256.f) - m * m + 1e-5f);
    }
    __syncthreads();
    for (int li = lane; li < 16 * 256; li += 32) {
        const int row = li >> 8, c = li & 255;
        float v = (H1[li] - MEANS[row]) * RSTDS[row] * g1[c] + be1[c];
        H1[li] = fmaxf(v, 0.f);
    }
    __syncthreads();

    // ---- layer 2: 256 -> 512 ----
    for (int nt = 0; nt < 32; ++nt) {
        const int n0 = nt * 16;
        v8f acc = {0.f,0.f,0.f,0.f,0.f,0.f,0.f,0.f};
        for (int k0 = 0; k0 < 256; k0 += 4) {
            const int kk = k0 + 2 * half;
            v2f a; a.x = H1[ml * 256 + kk]; a.y = H1[ml * 256 + kk + 1];
            v2f bf; bf.x = w2[kk * 512 + n0 + ml]; bf.y = w2[(kk + 1) * 512 + n0 + ml];
            acc = wmma4(a, bf, acc);
        }
        const float bbv = b2[n0 + ml];
        for (int r = 0; r < 8; ++r)
            H2[(r + 8 * half) * 512 + n0 + ml] = acc[r] + bbv;
    }
    __syncthreads();
    if (lane < 16) {
        float s = 0.f, s2 = 0.f;
        const float* row = &H2[lane * 512];
        for (int c = 0; c < 512; ++c) { float v = row[c]; s += v; s2 += v * v; }
        const float m = s * (1.f / 512.f);
        MEANS[lane] = m;
        RSTDS[lane] = rsqrtf(s2 * (1.f / 512.f) - m * m + 1e-5f);
    }
    __syncthreads();
    for (int li = lane; li < 16 * 512; li += 32) {
        const int row = li >> 9, c = li & 511;
        float v = (H2[li] - MEANS[row]) * RSTDS[row] * g2[c] + be2[c];
        H2[li] = fmaxf(v, 0.f);
    }
    __syncthreads();

    // ---- layer 3: 512 -> 512, masked store to output ----
    float* outp = isRef ? out_ref : out_tar;
    for (int nt = 0; nt < 32; ++nt) {
        const int n0 = nt * 16;
        v8f acc = {0.f,0.f,0.f,0.f,0.f,0.f,0.f,0.f};
        for (int k0 = 0; k0 < 512; k0 += 4) {
            const int kk = k0 + 2 * half;
            v2f a; a.x = H2[ml * 512 + kk]; a.y = H2[ml * 512 + kk + 1];
            v2f bf; bf.x = w3[kk * 512 + n0 + ml]; bf.y = w3[(kk + 1) * 512 + n0 + ml];
            acc = wmma4(a, bf, acc);
        }
        const float bbv = b3[n0 + ml];
        for (int r = 0; r < 8; ++r) {
            const int row = r + 8 * half;
            const float mv = (TYPEF[row] > 0.5f) ? 1.f : 0.f;
            outp[(size_t)(rbase + row) * 512 + n0 + ml] = (acc[r] + bbv) * mv;
        }
    }
}

// ---------------------------------------------------------------------------
// Inverse row norms (wave per row)
// ---------------------------------------------------------------------------
__global__ __launch_bounds__(256) void norms_kernel(
    const float* __restrict__ eref, const float* __restrict__ etar,
    float* __restrict__ rinv, float* __restrict__ tinv)
{
    const int wave = threadIdx.x >> 5, lane = threadIdx.x & 31;
    const int row = blockIdx.x * 8 + wave;           // 262144 rows total
    const bool isRef = row < BNR;
    const float* src = isRef ? (eref + (size_t)row * 512)
                             : (etar + (size_t)(row - BNR) * 512);
    float s = 0.f;
    for (int k = lane; k < 512; k += 32) { float v = src[k]; s += v * v; }
    for (int off = 16; off; off >>= 1) s += __shfl_xor(s, off, 32);
    if (lane == 0) {
        const float inv = 1.f / fmaxf(sqrtf(s), 1e-12f);
        if (isRef) rinv[row] = inv; else tinv[row - BNR] = inv;
    }
}

// ---------------------------------------------------------------------------
// Similarity: one wave per (batch, 16-row n-tile).
//   A rows (pre-scaled by rinv) staged in conflict-free paired layout.
//   B tiles (raw tar rows) fetched by the Tensor Data Mover into padded LDS,
//   double-buffered across m-tiles, synchronized with s_wait_tensorcnt.
//   Normalization by tinv[m] is applied to the WMMA result (identical math).
// ---------------------------------------------------------------------------
#define APAIR_OFF 0                 // 8192 floats  (32 KB)
#define BRAW_OFF  8192              // 2 x 8256 floats (2 x 33 KB, TDM-padded)
#define BRAW_SZ   8256

__global__ __launch_bounds__(32) void sim_kernel(
    const float* __restrict__ eref, const float* __restrict__ etar,
    const float* __restrict__ rinv, const float* __restrict__ tinv,
    float* __restrict__ sim_out, float* __restrict__ scores, int* __restrict__ idxout)
{
    __shared__ float SMEM[8192 + 2 * BRAW_SZ];   // 98816 B of the 320 KB WGP LDS

    const int b    = blockIdx.y;
    const int lane = threadIdx.x & 31;
    const int half = lane >> 4;
    const int ml   = lane & 15;
    const int n0   = blockIdx.x * 16;

    const float* R = eref + (size_t)b * 262144;
    const float* T = etar + (size_t)b * 262144;

    // kick off TDM for m-tile 0 while we stage A
    tdm_load_2d((unsigned)(BRAW_OFF * 4), (unsigned long long)(size_t)T, 512u, 16u, 512u);

    // stage normalized A rows: Apair[(k>>1)*32 + m*2 + (k&1)]  (8192 = 256*32)
    for (int li = lane; li < 8192; li += 32) {
        const int m = li >> 9, k = li & 511;
        const float v = R[(size_t)(n0 + m) * 512 + k] * rinv[b * 512 + n0 + m];
        SMEM[APAIR_OFF + ((k >> 1) << 5) + (m << 1) + (k & 1)] = v;
    }
    __syncthreads();

    float rowMax[8]; int rowIdx[8];
    for (int r = 0; r < 8; ++r) { rowMax[r] = -3.4e38f; rowIdx[r] = 0; }

    float* simb = sim_out + (size_t)b * 262144;
    const float tcol_base = 0.f; (void)tcol_base;

    for (int mt = 0; mt < 32; ++mt) {
        const int m0 = mt * 16;
        const int cur = mt & 1;

        if (mt + 1 < 32) {
            // prefetch next tar tile via TDM into the other buffer
            tdm_load_2d((unsigned)((BRAW_OFF + (1 - cur) * BRAW_SZ) * 4),
                        (unsigned long long)(size_t)(T + (size_t)(m0 + 16) * 512),
                        512u, 16u, 512u);
            __builtin_amdgcn_s_wait_tensorcnt(1);   // current tile resident
        } else {
            __builtin_amdgcn_s_wait_tensorcnt(0);
        }

        const float* Braw = &SMEM[BRAW_OFF + cur * BRAW_SZ];

        v8f acc = {0.f,0.f,0.f,0.f,0.f,0.f,0.f,0.f};
        for (int k0 = 0; k0 < 512; k0 += 4) {
            const int kk = k0 + 2 * half;
            v2f a = *(const v2f*)&SMEM[APAIR_OFF + ((kk >> 1) << 5) + (ml << 1)];
            // padded raw layout: word = ml*516 + kk + 2*(kk>=256); stride%64==4
            v2f bf = *(const v2f*)&Braw[ml * 516 + kk + ((kk >= 256) ? 2 : 0)];
            acc = wmma4(a, bf, acc);
        }

        const float tc = tinv[b * 512 + m0 + ml];   // per-column normalization
        for (int r = 0; r < 8; ++r) acc[r] *= tc;

        // write sim tile
        for (int r = 0; r < 8; ++r) {
            const int row = r + 8 * half;
            simb[(size_t)(n0 + row) * 512 + m0 + ml] = acc[r];
        }
        // running row max/argmax (first-max tie-break)
        for (int r = 0; r < 8; ++r) {
            float v = acc[r]; int ci = m0 + ml;
            for (int off = 1; off < 16; off <<= 1) {
                const float ov = __shfl_xor(v, off, 32);
                const int   oi = __shfl_xor(ci, off, 32);
                if (ov > v || (ov == v && oi < ci)) { v = ov; ci = oi; }
            }
            if (v > rowMax[r]) { rowMax[r] = v; rowIdx[r] = ci; }
        }
    }

    if (ml == 0) {  // lanes 0 and 16 cover rows 0..7 / 8..15
        for (int r = 0; r < 8; ++r) {
            const int row = n0 + r + 8 * half;
            scores[b * 512 + row] = rowMax[r];
            idxout[b * 512 + row] = rowIdx[r];
        }
    }
}

// ---------------------------------------------------------------------------
// Pool: per-batch mean of [ref | matched | score], padded to stride 1028
// ---------------------------------------------------------------------------
__global__ __launch_bounds__(256) void pool_kernel(
    const float* __restrict__ eref, const float* __restrict__ etar,
    const float* __restrict__ scores, const int* __restrict__ idxv,
    float* __restrict__ pooled)
{
    const int b = blockIdx.x, t = threadIdx.x;
    const float* Rb = eref + (size_t)b * 262144;
    const float* Tb = etar + (size_t)b * 262144;
    float sr0 = 0.f, sr1 = 0.f, sm0 = 0.f, sm1 = 0.f, ssc = 0.f;
    for (int n = 0; n < 512; ++n) {
        const float s = scores[b * 512 + n];
        const int id = idxv[b * 512 + n];
        const bool valid = s > 0.1f;
        const float* rr = Rb + (size_t)n * 512;
        if (n + 1 < 512) __builtin_prefetch(rr + 512 + t, 0, 1);
        sr0 += rr[t]; sr1 += rr[t + 256];
        if (valid) {
            const float* tr = Tb + (size_t)id * 512;
            sm0 += tr[t]; sm1 += tr[t + 256]; ssc += s;
        }
    }
    float* P = pooled + (size_t)b * 1028;
    const float inv = 1.f / 512.f;
    P[t] = sr0 * inv;  P[t + 256] = sr1 * inv;
    P[512 + t] = sm0 * inv;  P[512 + t + 256] = sm1 * inv;
    if (t == 0) { P[1024] = ssc * inv; P[1025] = 0.f; P[1026] = 0.f; P[1027] = 0.f; }
}

// ---------------------------------------------------------------------------
// Projection GEMM (pre-LN): pooled(256x1025) @ pw(1025x1024) + pb
// ---------------------------------------------------------------------------
__global__ __launch_bounds__(32) void proj_gemm_kernel(
    const float* __restrict__ pooled, const float* __restrict__ pw,
    const float* __restrict__ pb, float* __restrict__ pre)
{
    const int lane = threadIdx.x & 31, half = lane >> 4, ml = lane & 15;
    const int n0 = blockIdx.x * 16;   // 64
    const int r0 = blockIdx.y * 16;   // 16
    v8f acc = {0.f,0.f,0.f,0.f,0.f,0.f,0.f,0.f};
    for (int k0 = 0; k0 < 1024; k0 += 4) {
        const int kk = k0 + 2 * half;
        v2f a; a.x = pooled[(size_t)(r0 + ml) * 1028 + kk];
               a.y = pooled[(size_t)(r0 + ml) * 1028 + kk + 1];
        v2f bf; bf.x = pw[(size_t)kk * 1024 + n0 + ml];
                bf.y = pw[(size_t)(kk + 1) * 1024 + n0 + ml];
        acc = wmma4(a, bf, acc);
    }
    const float bl = pb[n0 + ml];
    const float bw = pw[(size_t)1024 * 1024 + n0 + ml];   // k = 1024 row
    for (int r = 0; r < 8; ++r) {
        const int row = r + 8 * half;
        const float alast = pooled[(size_t)(r0 + row) * 1028 + 1024];
        pre[(size_t)(r0 + row) * 1024 + n0 + ml] = acc[r] + bl + alast * bw;
    }
}

// ---------------------------------------------------------------------------
// LN + ReLU over 1024 per row -> proj output
// ---------------------------------------------------------------------------
__global__ __launch_bounds__(256) void proj_ln_kernel(
    const float* __restrict__ pre, const float* __restrict__ pg,
    const float* __restrict__ pbe, float* __restrict__ proj)
{
    __shared__ float red[256];
    const int b = blockIdx.x, t = threadIdx.x;
    const float* row = pre + (size_t)b * 1024;
    float s = 0.f, s2 = 0.f;
    for (int c = t; c < 1024; c += 256) { const float v = row[c]; s += v; s2 += v * v; }
    red[t] = s; __syncthreads();
    for (int off = 128; off; off >>= 1) { if (t < off) red[t] += red[t + off]; __syncthreads(); }
    const float mean = red[0] * (1.f / 1024.f);
    __syncthreads();
    red[t] = s2; __syncthreads();
    for (int off = 128; off; off >>= 1) { if (t < off) red[t] += red[t + off]; __syncthreads(); }
    const float var = red[0] * (1.f / 1024.f) - mean * mean;
    const float rstd = rsqrtf(var + 1e-5f);
    for (int c = t; c < 1024; c += 256) {
        const float v = (row[c] - mean) * rstd * pg[c] + pbe[c];
        proj[(size_t)b * 1024 + c] = fmaxf(v, 0.f);
    }
}

// ---------------------------------------------------------------------------
// Classifier head: logits = relu(proj@cw1+cb1) @ cw2 + cb2
// ---------------------------------------------------------------------------
__global__ __launch_bounds__(128) void head_kernel(
    const float* __restrict__ proj, const float* __restrict__ cw1,
    const float* __restrict__ cb1, const float* __restrict__ cw2,
    const float* __restrict__ cb2, float* __restrict__ logits)
{
    __shared__ float h[128];
    const int b = blockIdx.x, t = threadIdx.x;
    const float* row = proj + (size_t)b * 1024;
    float s = cb1[t];
    for (int k = 0; k < 1024; ++k) s += row[k] * cw1[k * 128 + t];
    h[t] = fmaxf(s, 0.f);
    __syncthreads();
    if (t < 4) {
        float o = cb2[t];
        for (int c = 0; c < 128; ++c) o += h[c] * cw2[c * 4 + t];
        logits[b * 4 + t] = o;
    }
}

extern "C" void kernel_launch(void* const* d_in, const int* in_sizes, int n_in,
                              void* d_out, int out_size, void* d_ws, size_t ws_size,
                              hipStream_t stream) {
    const float* refc = (const float*)d_in[0];
    const float* tarc = (const float*)d_in[1];
    const float* temb = (const float*)d_in[2];
    const float* w1  = (const float*)d_in[3];
    const float* b1  = (const float*)d_in[4];
    const float* g1  = (const float*)d_in[5];
    const float* be1 = (const float*)d_in[6];
    const float* w2  = (const float*)d_in[7];
    const float* b2  = (const float*)d_in[8];
    const float* g2  = (const float*)d_in[9];
    const float* be2 = (const float*)d_in[10];
    const float* w3  = (const float*)d_in[11];
    const float* b3  = (const float*)d_in[12];
    const float* pw  = (const float*)d_in[13];
    const float* pb  = (const float*)d_in[14];
    const float* pg  = (const float*)d_in[15];
    const float* pbe = (const float*)d_in[16];
    const float* cw1 = (const float*)d_in[17];
    const float* cb1 = (const float*)d_in[18];
    const float* cw2 = (const float*)d_in[19];
    const float* cb2 = (const float*)d_in[20];

    // outputs: proj | sim | logits | ref_encoded | tar_encoded (flat, in order)
    float* out    = (float*)d_out;
    float* proj   = out;
    float* sim    = out + 262144;
    float* logits = out + 67371008;
    float* eref   = out + 67372032;
    float* etar   = out + 134480896;

    // workspace layout (floats)
    float* ws     = (float*)d_ws;
    float* rinv   = ws;                      // 131072
    float* tinv   = ws + 131072;             // 131072
    float* scores = ws + 262144;             // 131072
    int*   idxv   = (int*)(ws + 393216);     // 131072
    float* pooled = ws + 524288;             // 256*1028
    float* pre    = ws + 787456;             // 256*1024

    encode_kernel<<<16384, 32, 0, stream>>>(refc, tarc, temb,
        w1, b1, g1, be1, w2, b2, g2, be2, w3, b3, eref, etar);
    norms_kernel<<<32768, 256, 0, stream>>>(eref, etar, rinv, tinv);
    sim_kernel<<<dim3(32, 256), 32, 0, stream>>>(eref, etar, rinv, tinv,
                                                 sim, scores, idxv);
    pool_kernel<<<256, 256, 0, stream>>>(eref, etar, scores, idxv, pooled);
    proj_gemm_kernel<<<dim3(64, 16), 32, 0, stream>>>(pooled, pw, pb, pre);
    proj_ln_kernel<<<256, 256, 0, stream>>>(pre, pg, pbe, proj);
    head_kernel<<<256, 128, 0, stream>>>(proj, cw1, cb1, cw2, cb2, logits);
}